// GNNFeatExtractor_46411416601225
// MI455X (gfx1250) — compile-verified
//
#include <hip/hip_runtime.h>

#define NNODES 100000
#define NEDGES 1600000

typedef __attribute__((ext_vector_type(2))) float v2f;
typedef __attribute__((ext_vector_type(8))) float v8f;

__device__ __forceinline__ void atomAddF32(float* p, float v) {
    // native global_atomic_add_f32 on gfx1250 (L2-resident scatter targets)
    unsafeAtomicAdd(p, v);
}

// ---------------- degree / norm ----------------
__global__ void deg_kernel(const int* __restrict__ col, const float* __restrict__ ew,
                           float* __restrict__ deg, int E) {
    int e = blockIdx.x * blockDim.x + threadIdx.x;
    if (e < E) atomAddF32(&deg[col[e]], ew[e]);
}

__global__ void dinv_kernel(float* __restrict__ deg, int n) {
    int i = blockIdx.x * blockDim.x + threadIdx.x;
    if (i < n) { float d = deg[i]; deg[i] = (d > 0.f) ? rsqrtf(d) : 0.f; }
}

__global__ void norm_kernel(const int* __restrict__ row, const int* __restrict__ col,
                            const float* __restrict__ ew, const float* __restrict__ dinv,
                            float* __restrict__ norm, int E) {
    int e = blockIdx.x * blockDim.x + threadIdx.x;
    if (e < E) norm[e] = dinv[row[e]] * ew[e] * dinv[col[e]];
}

// ---------------- edge propagation (scatter-add SpMM) ----------------
__global__ void prop3_kernel(const int* __restrict__ row, const int* __restrict__ col,
                             const float* __restrict__ norm, const float* __restrict__ hin,
                             float* __restrict__ hout, int E) {
    int e = blockIdx.x * blockDim.x + threadIdx.x;
    if (e >= E) return;
    int r = row[e], c = col[e];
    float w = norm[e];
    atomAddF32(&hout[c * 3 + 0], w * hin[r * 3 + 0]);
    atomAddF32(&hout[c * 3 + 1], w * hin[r * 3 + 1]);
    atomAddF32(&hout[c * 3 + 2], w * hin[r * 3 + 2]);
}

template <int D>
__global__ void prop_kernel(const int* __restrict__ row, const int* __restrict__ col,
                            const float* __restrict__ norm, const float* __restrict__ hin,
                            float* __restrict__ hout, int E) {
    constexpr int D4 = D / 4;
    int idx = blockIdx.x * blockDim.x + threadIdx.x;
    if (idx >= E * D4) return;
    int e = idx / D4;
    int c = idx % D4;
    int r = row[e], cl = col[e];
    float w = norm[e];
    const float4 v = *reinterpret_cast<const float4*>(hin + (size_t)r * D + 4 * c);
    float* dst = hout + (size_t)cl * D + 4 * c;
    atomAddF32(dst + 0, w * v.x);
    atomAddF32(dst + 1, w * v.y);
    atomAddF32(dst + 2, w * v.z);
    atomAddF32(dst + 3, w * v.w);
}

// ---------------- layer 1 (K=3, too small for WMMA) ----------------
__global__ void layer1_kernel(const float* __restrict__ x, const float* __restrict__ hop,
                              const float* __restrict__ W, const float* __restrict__ bias,
                              float* __restrict__ out) {
    int idx = blockIdx.x * blockDim.x + threadIdx.x;   // NNODES*32
    if (idx >= NNODES * 32) return;
    int i = idx >> 5, j = idx & 31;
    float acc = bias[j];
#pragma unroll
    for (int k = 0; k < 3; ++k) {
        acc += x[i * 3 + k]   * W[k * 32 + j];        // W[0]
        acc += hop[i * 3 + k] * W[96 + k * 32 + j];   // W[1]
    }
    out[idx] = fmaxf(acc, 0.f);
}

// ---------------- fused multi-term WMMA GEMM ----------------
// out[M x NOUT] = (Σ_t A_t[M x K] @ W[t][K x NOUT]) + bias, optional ReLU
// One block = 16 rows; wave w = 16-col tile. A-tile staged in LDS (also makes
// in-place A==out safe: all reads complete at __syncthreads before epilogue).
// V_WMMA_F32_16X16X4_F32 fragment maps (ISA 7.12.2, f32):
//   A/B: VGPR v, lane-half h -> K = v + 2h ;  C/D: VGPR r -> M = r + 8*half, N = lane&15
template <int K, int NT, int NOUT, bool RELU>
__global__ void tag_gemm_wmma(const float* __restrict__ a0, const float* __restrict__ a1,
                              const float* __restrict__ a2, const float* __restrict__ a3,
                              const float* __restrict__ W,   // [NT][K][NOUT]
                              const float* __restrict__ bias,
                              float* __restrict__ out) {
    constexpr int NTHREADS = (NOUT / 16) * 32;
    __shared__ float sA[16 * K];

    const int m0   = blockIdx.x * 16;
    const int tid  = threadIdx.x;
    const int lane = tid & 31;
    const int wave = tid >> 5;
    const int half = lane >> 4;
    const int l    = lane & 15;
    const int n0   = wave * 16;

    const float* As[4] = {a0, a1, a2, a3};
    v8f c = {};

#pragma unroll
    for (int t = 0; t < NT; ++t) {
        const float* A = As[t];
        for (int i = tid; i < 16 * K; i += NTHREADS) {
            int r = i / K, kk = i % K;
            sA[i] = A[(size_t)(m0 + r) * K + kk];
        }
        __syncthreads();
        const float* Wt = W + (size_t)t * K * NOUT;
#pragma unroll
        for (int kk = 0; kk < K; kk += 4) {
            int ka = kk + 2 * half;
            v2f a, b;
            a.x = sA[l * K + ka];
            a.y = sA[l * K + ka + 1];
            b.x = Wt[(size_t)ka * NOUT + n0 + l];
            b.y = Wt[(size_t)(ka + 1) * NOUT + n0 + l];
            c = __builtin_amdgcn_wmma_f32_16x16x4_f32(false, a, false, b,
                                                      (short)0, c, false, false);
        }
        __syncthreads();
    }

    float bv = bias[n0 + l];
#pragma unroll
    for (int r = 0; r < 8; ++r) {
        int rowi = m0 + r + 8 * half;
        float v = c[r] + bv;
        if (RELU) v = fmaxf(v, 0.f);
        out[(size_t)rowi * NOUT + n0 + l] = v;
    }
}

// ---------------- row L2 normalize (128 cols, one wave per row) ----------------
__global__ void l2norm_kernel(float* __restrict__ out, int n) {
    int node = blockIdx.x * (blockDim.x >> 5) + (threadIdx.x >> 5);
    int lane = threadIdx.x & 31;
    if (node >= n) return;
    float* p = out + (size_t)node * 128;
    float4 v = *reinterpret_cast<float4*>(p + lane * 4);
    float ss = v.x * v.x + v.y * v.y + v.z * v.z + v.w * v.w;
#pragma unroll
    for (int off = 16; off; off >>= 1) ss += __shfl_xor(ss, off, 32);
    float nrm = fmaxf(sqrtf(ss), 1e-12f);
    float inv = 1.f / nrm;
    v.x *= inv; v.y *= inv; v.z *= inv; v.w *= inv;
    *reinterpret_cast<float4*>(p + lane * 4) = v;
}

extern "C" void kernel_launch(void* const* d_in, const int* in_sizes, int n_in,
                              void* d_out, int out_size, void* d_ws, size_t ws_size,
                              hipStream_t stream) {
    const float* x    = (const float*)d_in[0];
    const int*   ei   = (const int*)d_in[1];
    const float* ew   = (const float*)d_in[2];
    const float* w1   = (const float*)d_in[3];
    const float* b1   = (const float*)d_in[4];
    const float* w2   = (const float*)d_in[5];
    const float* b2   = (const float*)d_in[6];
    const float* w3   = (const float*)d_in[7];
    const float* b3   = (const float*)d_in[8];
    const float* fcw  = (const float*)d_in[9];
    const float* fcb  = (const float*)d_in[10];
    float* out = (float*)d_out;

    const int N = NNODES, E = NEDGES;
    const int* row = ei;
    const int* col = ei + E;

    // workspace layout (floats); F1 (N*32) overlaps P2 (N*64): F1 dead before P2 use
    float* wsf  = (float*)d_ws;
    float* deg  = wsf;                       // N
    float* nrm  = deg + N;                   // E
    float* P0   = nrm + E;                   // N*64
    float* P1   = P0 + (size_t)N * 64;       // N*64
    float* F2   = P1 + (size_t)N * 64;       // N*64
    float* F1   = F2 + (size_t)N * 64;       // N*32 (within last N*64 slot)
    float* P2   = F1;                        // N*64 (reuses F1 region)

    const int TB = 256;
    const int gE  = (E + TB - 1) / TB;
    const int gN  = (N + TB - 1) / TB;
    const int gMT = N / 16;                  // 100000 % 16 == 0

    // gcn_norm
    hipMemsetAsync(deg, 0, (size_t)N * 4, stream);
    deg_kernel <<<gE, TB, 0, stream>>>(col, ew, deg, E);
    dinv_kernel<<<gN, TB, 0, stream>>>(deg, N);
    norm_kernel<<<gE, TB, 0, stream>>>(row, col, ew, deg, nrm, E);

    // layer 1: K=3 -> 32, 1 hop
    hipMemsetAsync(P0, 0, (size_t)N * 3 * 4, stream);
    prop3_kernel<<<gE, TB, 0, stream>>>(row, col, nrm, x, P0, E);
    layer1_kernel<<<(N * 32 + TB - 1) / TB, TB, 0, stream>>>(x, P0, w1, b1, F1);

    // layer 2: K=32 -> 64, 2 hops
    hipMemsetAsync(P0, 0, (size_t)N * 32 * 4, stream);
    prop_kernel<32><<<(E * 8 + TB - 1) / TB, TB, 0, stream>>>(row, col, nrm, F1, P0, E);
    hipMemsetAsync(P1, 0, (size_t)N * 32 * 4, stream);
    prop_kernel<32><<<(E * 8 + TB - 1) / TB, TB, 0, stream>>>(row, col, nrm, P0, P1, E);
    tag_gemm_wmma<32, 3, 64, true><<<gMT, 128, 0, stream>>>(F1, P0, P1, nullptr, w2, b2, F2);

    // layer 3: K=64 -> 128, 3 hops (output -> d_out)
    hipMemsetAsync(P0, 0, (size_t)N * 64 * 4, stream);
    prop_kernel<64><<<(E * 16 + TB - 1) / TB, TB, 0, stream>>>(row, col, nrm, F2, P0, E);
    hipMemsetAsync(P1, 0, (size_t)N * 64 * 4, stream);
    prop_kernel<64><<<(E * 16 + TB - 1) / TB, TB, 0, stream>>>(row, col, nrm, P0, P1, E);
    hipMemsetAsync(P2, 0, (size_t)N * 64 * 4, stream);
    prop_kernel<64><<<(E * 16 + TB - 1) / TB, TB, 0, stream>>>(row, col, nrm, P1, P2, E);
    tag_gemm_wmma<64, 4, 128, true><<<gMT, 256, 0, stream>>>(F2, P0, P1, P2, w3, b3, out);

    // fc (in-place on d_out; safe via LDS staging) + row L2 normalize
    tag_gemm_wmma<128, 1, 128, false><<<gMT, 256, 0, stream>>>(out, nullptr, nullptr, nullptr,
                                                               fcw, fcb, out);
    l2norm_kernel<<<(N + 7) / 8, 256, 0, stream>>>(out, N);
}